// DecodeLayer_39049842655282
// MI455X (gfx1250) — compile-verified
//
#include <hip/hip_runtime.h>
#include <stdint.h>

// ---------------- problem constants (match reference) ----------------
#define B_       128
#define N_       8732
#define C_       21
#define STRIDE_  25          // 4 loc + 21 class logits per anchor
#define MAX_OUT_ 5
#define NEG_     (-1e9f)
#define IOU_THR_ 0.5f
#define CONF_THR_ 0.5f

#define CHUNK_   256         // anchors staged per block (decode kernel)
#define TPB1_    256         // 8 wave32s
#define TPB2_    256

// ---------------- CDNA5 async global->LDS helpers ----------------
__device__ __forceinline__ void async_load_b128(uint32_t lds_addr,
                                                uint32_t byte_off,
                                                unsigned long long gbase) {
#if defined(__gfx1250__)
  // GVS form: mem_addr = SADDR(64) + VADDR(32) + inst_offset ; VDST = LDS byte addr
  asm volatile("global_load_async_to_lds_b128 %0, %1, %2 offset:0"
               :: "v"(lds_addr), "v"(byte_off), "s"(gbase)
               : "memory");
#else
  (void)lds_addr; (void)byte_off; (void)gbase;
#endif
}

__device__ __forceinline__ void async_wait_all() {
#if defined(__gfx1250__)
  asm volatile("s_wait_asynccnt 0" ::: "memory");
#endif
}

// ---------------- kernel 1: decode boxes + score/class per anchor ----------------
__global__ void __launch_bounds__(TPB1_)
decode_score_kernel(const float* __restrict__ logits,
                    const float* __restrict__ dbox,
                    float4* __restrict__ w_boxes,
                    float*  __restrict__ w_scores,
                    float*  __restrict__ w_classes) {
  __shared__ float tile[CHUNK_ * STRIDE_];   // 25,600 B

  const int chunksPerBatch = (N_ + CHUNK_ - 1) / CHUNK_;   // 35
  const int b     = blockIdx.x / chunksPerBatch;
  const int chunk = blockIdx.x % chunksPerBatch;
  const int a0    = chunk * CHUNK_;
  const int nAnc  = min(CHUNK_, N_ - a0);

  // ---- async-DMA this block's logits tile into LDS (all segments 16B aligned) ----
  const unsigned long long src =
      (unsigned long long)(uintptr_t)(logits + ((size_t)b * N_ + (size_t)a0) * STRIDE_);
  const int bytes = nAnc * STRIDE_ * 4;      // 25600 (full) or 2800 (tail): both %16==0
  for (int off = (int)threadIdx.x * 16; off < bytes; off += TPB1_ * 16) {
    uint32_t lds_addr = (uint32_t)(uintptr_t)((char*)tile + off);  // addr[31:0] = LDS offset
    async_load_b128(lds_addr, (uint32_t)off, src);
  }
  async_wait_all();
  __syncthreads();

  const int t = (int)threadIdx.x;
  if (t < nAnc) {
    const float* l = &tile[t * STRIDE_];

    // class max / argmax (first occurrence, matches jnp.argmax)
    float m = l[4];
    int cls = 0;
#pragma unroll
    for (int k = 1; k < C_; ++k) {
      float v = l[4 + k];
      if (v > m) { m = v; cls = k; }
    }
    // max(softmax) = 1 / sum(exp(x - max))
    float sum = 0.0f;
#pragma unroll
    for (int k = 0; k < C_; ++k) sum += __expf(l[4 + k] - m);
    const float score = 1.0f / sum;

    // decode against default box (corner format [ymin,xmin,ymax,xmax])
    const int a = a0 + t;
    const float4 db = ((const float4*)dbox)[a];
    const float cy = 0.5f * (db.z + db.x);
    const float cx = 0.5f * (db.w + db.y);
    const float h  = db.z - db.x;
    const float w  = db.w - db.y;
    const float ncy = l[0] * h + cy;
    const float ncx = l[1] * w + cx;
    const float nh  = __expf(l[2]) * h;
    const float nw  = __expf(l[3]) * w;

    float4 box;
    box.x = fminf(fmaxf(ncy - 0.5f * nh, 0.0f), 1.0f);
    box.y = fminf(fmaxf(ncx - 0.5f * nw, 0.0f), 1.0f);
    box.z = fminf(fmaxf(ncy + 0.5f * nh, 0.0f), 1.0f);
    box.w = fminf(fmaxf(ncx + 0.5f * nw, 0.0f), 1.0f);

    const size_t idx = (size_t)b * N_ + (size_t)a;
    w_boxes[idx]   = box;
    w_scores[idx]  = (cls != 0) ? score : NEG_;   // background -> -inf stand-in
    w_classes[idx] = (float)cls;
  }
}

// ---------------- kernel 2: per-image NMS (5 rounds) ----------------
__global__ void __launch_bounds__(TPB2_)
nms_kernel(const float4* __restrict__ w_boxes,
           const float*  __restrict__ w_scores,
           const float*  __restrict__ w_classes,
           float* __restrict__ out) {
  __shared__ float  s_scores[N_];      // 34,928 B (fits easily in 320KB WGP LDS)
  __shared__ float  red_s[TPB2_];
  __shared__ int    red_i[TPB2_];
  __shared__ float4 s_pick;

  const int b = (int)blockIdx.x;
  const float*  sc = w_scores + (size_t)b * N_;
  const float4* bx = w_boxes  + (size_t)b * N_;

  for (int j = (int)threadIdx.x; j < N_; j += TPB2_) s_scores[j] = sc[j];
  __syncthreads();

  for (int it = 0; it < MAX_OUT_; ++it) {
    // --- block-wide argmax with first-occurrence tie-break ---
    float best = -INFINITY;
    int   bi   = N_;
    for (int j = (int)threadIdx.x; j < N_; j += TPB2_) {
      float v = s_scores[j];
      if (v > best) { best = v; bi = j; }   // ascending j: strict > keeps earliest
    }
    red_s[threadIdx.x] = best;
    red_i[threadIdx.x] = bi;
    __syncthreads();
    for (int s = TPB2_ / 2; s > 0; s >>= 1) {
      if ((int)threadIdx.x < s) {
        float vo = red_s[threadIdx.x + s];
        int   io = red_i[threadIdx.x + s];
        float vm = red_s[threadIdx.x];
        int   im = red_i[threadIdx.x];
        if (vo > vm || (vo == vm && io < im)) {
          red_s[threadIdx.x] = vo;
          red_i[threadIdx.x] = io;
        }
      }
      __syncthreads();
    }

    if (threadIdx.x == 0) {
      const int   i = red_i[0];
      const float v = red_s[0];
      const float4 box = bx[i];
      s_pick = box;
      const bool valid = (v > CONF_THR_);    // TF NMS score_threshold
      float* o = out + ((size_t)b * MAX_OUT_ + (size_t)it) * 6;
      o[0] = valid ? box.x : 0.0f;
      o[1] = valid ? box.y : 0.0f;
      o[2] = valid ? box.z : 0.0f;
      o[3] = valid ? box.w : 0.0f;
      o[4] = valid ? w_classes[(size_t)b * N_ + (size_t)i] : 0.0f;
      o[5] = valid ? v : 0.0f;
    }
    __syncthreads();

    // --- suppress everything with IoU > 0.5 vs picked box (suppresses pick too) ---
    const float4 pb = s_pick;
    const float a1 = (pb.z - pb.x) * (pb.w - pb.y);
    for (int j = (int)threadIdx.x; j < N_; j += TPB2_) {
      const float4 cb = bx[j];        // L2-resident on re-reads (192MB L2)
      const float ty  = fmaxf(pb.x, cb.x);
      const float tx  = fmaxf(pb.y, cb.y);
      const float by  = fminf(pb.z, cb.z);
      const float bxe = fminf(pb.w, cb.w);
      const float hh  = fmaxf(by - ty, 0.0f);
      const float ww  = fmaxf(bxe - tx, 0.0f);
      const float inter = hh * ww;
      const float a2 = (cb.z - cb.x) * (cb.w - cb.y);
      const float iou = inter / (a1 + a2 - inter + 1e-12f);
      if (iou > IOU_THR_) s_scores[j] = NEG_;
    }
    __syncthreads();
  }
}

// ---------------- launch ----------------
extern "C" void kernel_launch(void* const* d_in, const int* in_sizes, int n_in,
                              void* d_out, int out_size, void* d_ws, size_t ws_size,
                              hipStream_t stream) {
  (void)in_sizes; (void)n_in; (void)out_size; (void)ws_size;
  const float* logits = (const float*)d_in[0];   // [128, 8732, 25] f32
  const float* dbox   = (const float*)d_in[1];   // [8732, 4] f32

  char* ws = (char*)d_ws;
  const size_t bn = (size_t)B_ * (size_t)N_;
  float4* w_boxes   = (float4*)ws;                                       // 17.9 MB
  float*  w_scores  = (float*)(ws + bn * sizeof(float4));                //  4.5 MB
  float*  w_classes = (float*)(ws + bn * sizeof(float4) + bn * sizeof(float)); // 4.5 MB

  const int chunksPerBatch = (N_ + CHUNK_ - 1) / CHUNK_;   // 35
  decode_score_kernel<<<dim3(B_ * chunksPerBatch), dim3(TPB1_), 0, stream>>>(
      logits, dbox, w_boxes, w_scores, w_classes);
  nms_kernel<<<dim3(B_), dim3(TPB2_), 0, stream>>>(
      w_boxes, w_scores, w_classes, (float*)d_out);
}